// ReformerSeq2Seq_1494648619503
// MI455X (gfx1250) — compile-verified
//
#include <hip/hip_runtime.h>
#include <math.h>

// ---- problem constants (from reference) ----
#define BB 2
#define SS 4096
#define HID 512
#define HH 8
#define DD 64
#define CHUNK 64
#define NHASH 4
#define NBKT 128
#define NCHUNK 256           // NH*S/CHUNK
#define NBINS 512            // NHASH*NBKT
#define EXPN 16384           // NHASH*SS
#define SELF_MASK (-1.0e5f)

typedef float v2f __attribute__((ext_vector_type(2)));
typedef float v8f __attribute__((ext_vector_type(8)));

// async global->LDS copy of 16B for this lane; LDS dest from generic ptr low bits
__device__ __forceinline__ void async_b128(const void* lds_dst, const void* gsrc) {
  unsigned lds = (unsigned)(size_t)lds_dst;               // LDS byte offset
  unsigned long long ga = (unsigned long long)(size_t)gsrc;
  asm volatile("global_load_async_to_lds_b128 %0, %1, off"
               :: "v"(lds), "v"(ga) : "memory");
}
__device__ __forceinline__ void wait_async0() {
  asm volatile("s_wait_asynccnt 0x0" ::: "memory");
}

// =====================================================================
// K1: q/k/v projections.  GEMM M=8192 (B*S), K=512, N=512.
// grid = (N/64, M/64, 3), block = 128 (4 waves). 64x64 tile per block,
// each wave owns a 16-row stripe (4 C tiles), fp32 WMMA 16x16x4.
// Operand slabs staged via GLOBAL_LOAD_ASYNC_TO_LDS_B128 (ASYNCcnt).
// Output written split-head: O[((b*H+h)*S+s)*D+d], h == blockIdx.x.
// =====================================================================
__global__ __launch_bounds__(128) void k_proj(
    const float* __restrict__ ds, const float* __restrict__ hs,
    const float* __restrict__ wqk, const float* __restrict__ wv,
    float* __restrict__ q, float* __restrict__ k, float* __restrict__ v) {
  const int which = blockIdx.z;
  const float* A = (which == 0) ? ds : hs;
  const float* W = (which == 2) ? wv : wqk;
  float* O = (which == 0) ? q : ((which == 1) ? k : v);
  const int n0 = blockIdx.x * 64;
  const int m0 = blockIdx.y * 64;

  __shared__ __align__(16) float As[64 * 68];  // [row][k], pad 68
  __shared__ __align__(16) float Bs[64 * 72];  // [k][col], pad 72

  const int t = threadIdx.x;
  const int lane = t & 31;
  const int wv_ = t >> 5;
  const int lh = lane & 15;
  const int kb = (lane >> 4) * 2;

  v8f acc[4] = {};
  for (int k0 = 0; k0 < HID; k0 += 64) {
#pragma unroll
    for (int i = 0; i < 8; ++i) {
      int lin = i * 512 + t * 4;
      int r = lin >> 6, c = lin & 63;
      async_b128(&As[r * 68 + c], A + (size_t)(m0 + r) * HID + k0 + c);
      async_b128(&Bs[r * 72 + c], W + (size_t)(k0 + r) * HID + n0 + c);
    }
    wait_async0();
    __syncthreads();
    for (int kk = 0; kk < 64; kk += 4) {
      v2f a;
      a.x = As[(wv_ * 16 + lh) * 68 + kk + kb];
      a.y = As[(wv_ * 16 + lh) * 68 + kk + kb + 1];
#pragma unroll
      for (int nt = 0; nt < 4; ++nt) {
        v2f b;
        b.x = Bs[(kk + kb) * 72 + nt * 16 + lh];
        b.y = Bs[(kk + kb + 1) * 72 + nt * 16 + lh];
        acc[nt] = __builtin_amdgcn_wmma_f32_16x16x4_f32(
            false, a, false, b, (short)0, acc[nt], false, false);
      }
    }
    __syncthreads();
  }
  const int h = blockIdx.x;
  const int rbase = (lane < 16) ? 0 : 8;
#pragma unroll
  for (int nt = 0; nt < 4; ++nt) {
#pragma unroll
    for (int r = 0; r < 8; ++r) {
      int row = m0 + wv_ * 16 + r + rbase;
      int col = nt * 16 + lh;
      int b = row >> 12;
      int s = row & (SS - 1);
      O[((size_t)((b * HH + h) * SS + s)) * DD + col] = acc[nt][r];
    }
  }
}

// =====================================================================
// K2: LSH hash. thread per (b,h,n,s). rv[r] = sum_d x[d]*rot[h,d,n,r];
// bucket = first-argmax over [rv, -rv] (128) + n*NBKT.
// grid = (1024, 2), block = 256; grid.y selects q vs k.
// =====================================================================
__global__ __launch_bounds__(256) void k_hash(
    const float* __restrict__ q, const float* __restrict__ k,
    const float* __restrict__ rot, int* __restrict__ qb, int* __restrict__ kb) {
  int tid = blockIdx.x * 256 + threadIdx.x;
  const float* X = (blockIdx.y == 0) ? q : k;
  int* OB = (blockIdx.y == 0) ? qb : kb;
  int s = tid & (SS - 1);
  int n = (tid >> 12) & (NHASH - 1);
  int h = (tid >> 14) & (HH - 1);
  int b = tid >> 17;
  const float* xr = X + ((size_t)((b * HH + h) * SS + s)) * DD;

  float acc[64];
#pragma unroll
  for (int r = 0; r < 64; ++r) acc[r] = 0.f;
  for (int d = 0; d < DD; ++d) {
    float xv = xr[d];
    const float* rr = rot + ((size_t)((h * DD + d) * NHASH + n)) * 64;
#pragma unroll
    for (int i = 0; i < 16; ++i) {
      float4 r4 = *(const float4*)(rr + i * 4);
      acc[i * 4 + 0] += xv * r4.x;
      acc[i * 4 + 1] += xv * r4.y;
      acc[i * 4 + 2] += xv * r4.z;
      acc[i * 4 + 3] += xv * r4.w;
    }
  }
  float best = acc[0];
  int bi = 0;
#pragma unroll
  for (int j = 1; j < 64; ++j)
    if (acc[j] > best) { best = acc[j]; bi = j; }
#pragma unroll
  for (int j = 0; j < 64; ++j) {
    float vv = -acc[j];
    if (vv > best) { best = vv; bi = 64 + j; }
  }
  OB[(size_t)(b * HH + h) * EXPN + n * SS + s] = bi + n * NBKT;
}

// =====================================================================
// K3: stable counting sort by bucket (== argsort of S*bucket + i%S).
// grid = (16, 2) [bh, which], block = 1024. LDS < 64KB (16-bit buckets).
// which==0: write sq%S.  which==1: write sk%S and undo_k[i]=rank.
// =====================================================================
__global__ __launch_bounds__(1024) void k_sort(
    const int* __restrict__ qb, const int* __restrict__ kb,
    int* __restrict__ sqm, int* __restrict__ skm, int* __restrict__ undo) {
  const int bh = blockIdx.x;
  const int which = blockIdx.y;
  const int* src = which ? kb : qb;
  __shared__ unsigned short bkt[EXPN];   // 32KB
  __shared__ int cnt[NBINS * 2];
  __shared__ int start[NBINS * 2];
  const int t = threadIdx.x;
  const size_t base = (size_t)bh * EXPN;

  for (int i = t; i < EXPN; i += 1024) bkt[i] = (unsigned short)src[base + i];
  cnt[t] = 0;
  __syncthreads();
  for (int i = t; i < EXPN; i += 1024)
    atomicAdd(&cnt[((int)bkt[i] << 1) | (i >> 13)], 1);
  __syncthreads();
  if (t == 0) {
    int run = 0;
    for (int j = 0; j < NBINS * 2; ++j) { start[j] = run; run += cnt[j]; }
  }
  __syncthreads();
  const int bucket = t >> 1;
  const int half = t & 1;
  int off = start[t];
  const int i0 = half << 13;
  for (int i = i0; i < i0 + 8192; ++i) {
    if ((int)bkt[i] == bucket) {
      if (which == 0) {
        sqm[base + off] = i & (SS - 1);
      } else {
        skm[base + off] = i & (SS - 1);
        undo[base + i] = off;
      }
      ++off;
    }
  }
}

// =====================================================================
// K4: chunked attention. block per (chunk, bh), 128 threads (4 waves).
// dots = q[64x64] @ k_norm^T[64x128] (fp32 WMMA); self-mask on q-sorted
// indices; row softmax via 16-lane shfl reductions; out = P @ V with V
// B-fragments gathered from L2 via skidx. LDS kept < 64KB (ps overlays ks).
// q tile staged via async global->LDS; k tile via VGPRs (needs row norm).
// =====================================================================
__global__ __launch_bounds__(128) void k_attn(
    const float* __restrict__ q, const float* __restrict__ k,
    const float* __restrict__ v, const int* __restrict__ sqm,
    const int* __restrict__ skm, float* __restrict__ outS,
    float* __restrict__ lseS) {
  const int c = blockIdx.x;
  const int bh = blockIdx.y;
  __shared__ __align__(16) float qs[64 * 68];     // 17.4KB  [qrow][d]
  __shared__ __align__(16) float ksps[128 * 68];  // 34.8KB  ks[krow][d] -> ps[qrow][132]
  __shared__ int qidx[64];
  __shared__ int kqidx[128];
  __shared__ int skidx[128];

  const int t = threadIdx.x;
  const int lane = t & 31;
  const int wv_ = t >> 5;
  const int lh = lane & 15;
  const int kb = (lane >> 4) * 2;
  const size_t sbase = (size_t)bh * EXPN;
  const int cprev = (c + NCHUNK - 1) & (NCHUNK - 1);

  // ---- gather: q rows async (t<64), k rows w/ normalization (all 128) ----
  if (t < 64) {
    int qrow = sqm[sbase + c * 64 + t];
    qidx[t] = qrow;
    const float* qp = q + ((size_t)bh * SS + qrow) * DD;
#pragma unroll
    for (int i = 0; i < 16; ++i)
      async_b128(&qs[t * 68 + i * 4], qp + i * 4);
  }
  {
    int j = (t < 64) ? (cprev * 64 + t) : (c * 64 + (t - 64));
    kqidx[t] = sqm[sbase + j];
    int kr = skm[sbase + j];
    skidx[t] = kr;
    const float* kp = k + ((size_t)bh * SS + kr) * DD;
    float tmp[64];
    float ssq = 0.f;
#pragma unroll
    for (int i = 0; i < 16; ++i) {
      float4 kv = *(const float4*)(kp + i * 4);
      tmp[i * 4 + 0] = kv.x; tmp[i * 4 + 1] = kv.y;
      tmp[i * 4 + 2] = kv.z; tmp[i * 4 + 3] = kv.w;
      ssq += kv.x * kv.x + kv.y * kv.y + kv.z * kv.z + kv.w * kv.w;
    }
    float scale = rsqrtf(ssq * (1.0f / DD) + 1e-6f) * 0.125f;  // 1/sqrt(D)=0.125
#pragma unroll
    for (int i = 0; i < 64; ++i) ksps[t * 68 + i] = tmp[i] * scale;
  }
  wait_async0();
  __syncthreads();

  // ---- dots[64x128] = q @ k^T ----
  v8f acc[8] = {};
  for (int kk = 0; kk < 64; kk += 4) {
    v2f a;
    a.x = qs[(wv_ * 16 + lh) * 68 + kk + kb];
    a.y = qs[(wv_ * 16 + lh) * 68 + kk + kb + 1];
#pragma unroll
    for (int nt = 0; nt < 8; ++nt) {
      v2f b;
      b.x = ksps[(nt * 16 + lh) * 68 + kk + kb];
      b.y = ksps[(nt * 16 + lh) * 68 + kk + kb + 1];
      acc[nt] = __builtin_amdgcn_wmma_f32_16x16x4_f32(
          false, a, false, b, (short)0, acc[nt], false, false);
    }
  }
  __syncthreads();  // done reading ks; ksps becomes ps[64][132]

  // ---- mask + softmax (rows of this wave's stripe) ----
  const int rbase = (lane < 16) ? 0 : 8;
#pragma unroll
  for (int r = 0; r < 8; ++r) {
    int row = wv_ * 16 + r + rbase;
    int qi = qidx[row];
    float d0[8];
    float m = -3.0e38f;
#pragma unroll
    for (int nt = 0; nt < 8; ++nt) {
      float dd = acc[nt][r];
      if (kqidx[nt * 16 + lh] == qi) dd = SELF_MASK;
      d0[nt] = dd;
      m = fmaxf(m, dd);
    }
    for (int x = 1; x < 16; x <<= 1) m = fmaxf(m, __shfl_xor(m, x, 32));
    float sum = 0.f;
#pragma unroll
    for (int nt = 0; nt < 8; ++nt) { d0[nt] = expf(d0[nt] - m); sum += d0[nt]; }
    for (int x = 1; x < 16; x <<= 1) sum += __shfl_xor(sum, x, 32);
    float inv = 1.0f / sum;
#pragma unroll
    for (int nt = 0; nt < 8; ++nt)
      ksps[row * 132 + nt * 16 + lh] = d0[nt] * inv;
    if (lh == 0) lseS[sbase + c * 64 + row] = m + logf(sum);
  }
  __syncthreads();

  // ---- out[64x64] = P[64x128] @ V[128x64]; V frags gathered from L2 ----
  const float* vb = v + (size_t)bh * SS * DD;
  v8f acc2[4] = {};
  for (int kk = 0; kk < 128; kk += 4) {
    v2f a;
    a.x = ksps[(wv_ * 16 + lh) * 132 + kk + kb];
    a.y = ksps[(wv_ * 16 + lh) * 132 + kk + kb + 1];
    int r0 = skidx[kk + kb];
    int r1 = skidx[kk + kb + 1];
#pragma unroll
    for (int nt = 0; nt < 4; ++nt) {
      int col = nt * 16 + lh;
      v2f b;
      b.x = vb[(size_t)r0 * DD + col];
      b.y = vb[(size_t)r1 * DD + col];
      acc2[nt] = __builtin_amdgcn_wmma_f32_16x16x4_f32(
          false, a, false, b, (short)0, acc2[nt], false, false);
    }
  }
#pragma unroll
  for (int nt = 0; nt < 4; ++nt) {
#pragma unroll
    for (int r = 0; r < 8; ++r) {
      int row = wv_ * 16 + r + rbase;
      outS[(sbase + c * 64 + row) * DD + nt * 16 + lh] = acc2[nt][r];
    }
  }
}

// =====================================================================
// K5: reverse-sort gather (undo_k) + logsumexp combine of NH rounds.
// thread per (b,h,s,d4): 1,048,576 threads, float4 I/O.
// out[b,s,h*D+d] fp32.
// =====================================================================
__global__ __launch_bounds__(256) void k_combine(
    const float* __restrict__ outS, const float* __restrict__ lseS,
    const int* __restrict__ undo, float* __restrict__ out) {
  int tid = blockIdx.x * 256 + threadIdx.x;
  int d4 = tid & 15;
  int s = (tid >> 4) & (SS - 1);
  int h = (tid >> 16) & (HH - 1);
  int b = tid >> 19;
  size_t base = (size_t)(b * HH + h) * EXPN;
  int rr[NHASH];
  float l[NHASH];
  float m = -3.0e38f;
#pragma unroll
  for (int n = 0; n < NHASH; ++n) {
    rr[n] = undo[base + n * SS + s];
    l[n] = lseS[base + rr[n]];
    m = fmaxf(m, l[n]);
  }
  float Z = 0.f;
#pragma unroll
  for (int n = 0; n < NHASH; ++n) { l[n] = expf(l[n] - m); Z += l[n]; }
  float invZ = 1.0f / Z;
  float4 o = {0.f, 0.f, 0.f, 0.f};
#pragma unroll
  for (int n = 0; n < NHASH; ++n) {
    float w = l[n] * invZ;
    float4 ov = *(const float4*)(outS + (base + rr[n]) * DD + d4 * 4);
    o.x += w * ov.x; o.y += w * ov.y; o.z += w * ov.z; o.w += w * ov.w;
  }
  *(float4*)(out + ((size_t)(b * SS + s)) * (HH * DD) + h * DD + d4 * 4) = o;
}

// =====================================================================
// launch
// =====================================================================
extern "C" void kernel_launch(void* const* d_in, const int* in_sizes, int n_in,
                              void* d_out, int out_size, void* d_ws,
                              size_t ws_size, hipStream_t stream) {
  (void)in_sizes; (void)n_in; (void)out_size; (void)ws_size;
  const float* ds = (const float*)d_in[0];   // decoder_states [B,S,HID]
  const float* hs = (const float*)d_in[1];   // hidden_states  [B,S,HID]
  const float* wqk = (const float*)d_in[2];  // [HID, H*D]
  const float* wv = (const float*)d_in[3];   // [HID, H*D]
  const float* rot = (const float*)d_in[4];  // [H, D, NH, NB/2]

  char* ws = (char*)d_ws;
  const size_t QKV = (size_t)BB * HH * SS * DD * sizeof(float);  // 16MB
  const size_t IDX = (size_t)BB * HH * EXPN * sizeof(int);       // 1MB
  float* q = (float*)(ws);
  float* k = (float*)(ws + QKV);
  float* v = (float*)(ws + 2 * QKV);
  int* qb = (int*)(ws + 3 * QKV);
  int* kb = (int*)(ws + 3 * QKV + IDX);
  int* sqm = (int*)(ws + 3 * QKV + 2 * IDX);
  int* skm = (int*)(ws + 3 * QKV + 3 * IDX);
  int* undo = (int*)(ws + 3 * QKV + 4 * IDX);
  float* outS = (float*)(ws + 3 * QKV + 5 * IDX);                // 64MB
  float* lseS = (float*)(ws + 3 * QKV + 5 * IDX +
                         (size_t)BB * HH * EXPN * DD * sizeof(float));
  float* out = (float*)d_out;

  k_proj<<<dim3(HID / 64, (BB * SS) / 64, 3), 128, 0, stream>>>(
      ds, hs, wqk, wv, q, k, v);
  k_hash<<<dim3((BB * HH * NHASH * SS) / 256, 2), 256, 0, stream>>>(
      q, k, rot, qb, kb);
  k_sort<<<dim3(BB * HH, 2), 1024, 0, stream>>>(qb, kb, sqm, skm, undo);
  k_attn<<<dim3(NCHUNK, BB * HH), 128, 0, stream>>>(q, k, v, sqm, skm, outS,
                                                    lseS);
  k_combine<<<(BB * HH * SS * (DD / 4)) / 256, 256, 0, stream>>>(outS, lseS,
                                                                 undo, out);
}